// Encoder1_77618648973415
// MI455X (gfx1250) — compile-verified
//
#include <hip/hip_runtime.h>
#include <hip/hip_bf16.h>

typedef __attribute__((ext_vector_type(2))) float v2f;
typedef __attribute__((ext_vector_type(8))) float v8f;

#define DFEAT 128

// ---------------- zero fill (u32 semantics; 0 bit-pattern == 0.0f) ----------------
__global__ void zero_kernel(unsigned* __restrict__ p, size_t n) {
    size_t i = (size_t)blockIdx.x * blockDim.x + threadIdx.x;
    size_t stride = (size_t)gridDim.x * blockDim.x;
    for (; i < n; i += stride) p[i] = 0u;
}

// ---------------- degree counting ----------------
__global__ void degree_kernel(const int* __restrict__ src, const int* __restrict__ dst,
                              unsigned* __restrict__ degO, unsigned* __restrict__ degI, int nE) {
    int e = blockIdx.x * blockDim.x + threadIdx.x;
    if (e >= nE) return;
    atomicAdd(&degO[src[e]], 1u);
    atomicAdd(&degI[dst[e]], 1u);
}

// ---------------- exclusive scan of degI -> rowPtr (3 small kernels) ----------------
__global__ void scanA_kernel(const unsigned* __restrict__ degI, unsigned* __restrict__ rowPtr,
                             unsigned* __restrict__ blockSum, int n) {
    __shared__ unsigned sh[256];
    int t = threadIdx.x;
    int i = blockIdx.x * 256 + t;
    unsigned v = (i < n) ? degI[i] : 0u;
    sh[t] = v;
    __syncthreads();
#pragma unroll
    for (int off = 1; off < 256; off <<= 1) {
        unsigned add = (t >= off) ? sh[t - off] : 0u;
        __syncthreads();
        sh[t] += add;
        __syncthreads();
    }
    if (i < n) rowPtr[i] = sh[t] - v;          // exclusive within block
    if (t == 255) blockSum[blockIdx.x] = sh[255];
}

__global__ void scanB_kernel(unsigned* __restrict__ blockSum, int nb) {
    __shared__ unsigned sh[1024];
    int t = threadIdx.x;
    unsigned v = (t < nb) ? blockSum[t] : 0u;
    sh[t] = v;
    __syncthreads();
#pragma unroll
    for (int off = 1; off < 1024; off <<= 1) {
        unsigned add = (t >= off) ? sh[t - off] : 0u;
        __syncthreads();
        sh[t] += add;
        __syncthreads();
    }
    if (t < nb) blockSum[t] = sh[t] - v;       // exclusive block offsets
}

__global__ void scanC_kernel(unsigned* __restrict__ rowPtr, const unsigned* __restrict__ blockSum,
                             int n) {
    int i = blockIdx.x * 256 + threadIdx.x;
    if (i < n) rowPtr[i] += blockSum[blockIdx.x];
}

// ---------------- bucket fill: per-dst edge lists (int atomics only) ----------------
__global__ void fill_kernel(const int* __restrict__ src, const int* __restrict__ dst,
                            const unsigned* __restrict__ rowPtr, unsigned* __restrict__ cursor,
                            int* __restrict__ eSrc, int nE) {
    int e = blockIdx.x * blockDim.x + threadIdx.x;
    if (e >= nE) return;
    int d = dst[e];
    unsigned pos = atomicAdd(&cursor[d], 1u);
    eSrc[rowPtr[d] + pos] = src[e];
}

// ---------------- repack W into K-pair-interleaved layout for b64 B-loads ----------------
// Wp[(k>>1)*256 + col*2 + (k&1)] = W[k*128 + col]
__global__ void repack_kernel(const float* __restrict__ W, float* __restrict__ Wp) {
    int idx = blockIdx.x * 256 + threadIdx.x;       // 0..16383
    int k = idx >> 7, col = idx & 127;
    Wp[((k >> 1) << 8) + (col << 1) + (k & 1)] = W[idx];
}

// ---------------- x = (heat * rsqrt(deg_out)) @ W  via fp32 WMMA ----------------
__global__ void gemm_wmma_kernel(const float* __restrict__ heat,
                                 const float* __restrict__ Wp,
                                 const unsigned* __restrict__ degO,
                                 float* __restrict__ x, int n) {
    int wave = (blockIdx.x * blockDim.x + threadIdx.x) >> 5;
    int lane = threadIdx.x & 31;
    int row0 = wave * 16;
    if (row0 >= n) return;                  // wave-uniform: EXEC stays all-1s

    int m  = lane & 15;                     // M index in tile (and N index for B/D)
    int hi = lane >> 4;                     // K-half selector
    int row = row0 + m;

    float rs = rsqrtf(fmaxf((float)degO[row], 1.0f));

    v8f acc[8];
#pragma unroll
    for (int t = 0; t < 8; ++t) acc[t] = (v8f)(0.0f);

    const float* Arow = heat + (size_t)row * DFEAT;

    for (int k = 0; k < DFEAT; k += 4) {
        int kb = k + 2 * hi;                // lanes 0-15: K=k,k+1 ; lanes 16-31: K=k+2,k+3
        v2f a;
        a.x = Arow[kb]     * rs;
        a.y = Arow[kb + 1] * rs;
        const float* WpK = Wp + ((size_t)((k >> 1) + hi) << 8);  // pair row, 256 floats
#pragma unroll
        for (int t = 0; t < 8; ++t) {
            v2f b = *(const v2f*)(WpK + ((t * 16 + m) << 1));    // contiguous 8B -> b64
            acc[t] = __builtin_amdgcn_wmma_f32_16x16x4_f32(
                false, a, false, b, (short)0, acc[t], false, false);
        }
    }

    // D layout: VGPR r holds D[M = r + 8*hi][N = m]
#pragma unroll
    for (int t = 0; t < 8; ++t) {
#pragma unroll
        for (int r = 0; r < 8; ++r) {
            x[(size_t)(row0 + r + 8 * hi) * DFEAT + t * 16 + m] = acc[t][r];
        }
    }
}

// ---------------- CSR gather: h[r] = prelu(sum_{e in in(r)} x[src_e] * rsqrt(deg_in) + b) ----
// One wave per destination row; lanes hold a float4 column slice. No float atomics.
__global__ void gather_kernel(const float* __restrict__ x, const int* __restrict__ eSrc,
                              const unsigned* __restrict__ rowPtr, const unsigned* __restrict__ degI,
                              const float* __restrict__ b, const float* __restrict__ a1,
                              float* __restrict__ out, int n) {
    int wave = (blockIdx.x * blockDim.x + threadIdx.x) >> 5;
    int lane = threadIdx.x & 31;
    int r = __builtin_amdgcn_readfirstlane(wave);   // force scalar row index
    if (r >= n) return;

    unsigned start = rowPtr[r];
    unsigned cnt = degI[r];
    const int c4 = lane * 4;

    float4 acc = make_float4(0.f, 0.f, 0.f, 0.f);
    unsigned i = 0;
    for (; i + 4 <= cnt; i += 4) {                  // 4-wide MLP
        int s0 = eSrc[start + i + 0];
        int s1 = eSrc[start + i + 1];
        int s2 = eSrc[start + i + 2];
        int s3 = eSrc[start + i + 3];
        float4 v0 = *(const float4*)(x + (size_t)s0 * DFEAT + c4);
        float4 v1 = *(const float4*)(x + (size_t)s1 * DFEAT + c4);
        float4 v2 = *(const float4*)(x + (size_t)s2 * DFEAT + c4);
        float4 v3 = *(const float4*)(x + (size_t)s3 * DFEAT + c4);
        acc.x += v0.x + v1.x + v2.x + v3.x;
        acc.y += v0.y + v1.y + v2.y + v3.y;
        acc.z += v0.z + v1.z + v2.z + v3.z;
        acc.w += v0.w + v1.w + v2.w + v3.w;
    }
    for (; i < cnt; ++i) {
        int s = eSrc[start + i];
        float4 v = *(const float4*)(x + (size_t)s * DFEAT + c4);
        acc.x += v.x; acc.y += v.y; acc.z += v.z; acc.w += v.w;
    }

    float ri = rsqrtf(fmaxf((float)cnt, 1.0f));
    float al = a1[0];
    float4 bv = *(const float4*)(b + c4);
    float4 h;
    h.x = acc.x * ri + bv.x; h.x = (h.x > 0.f) ? h.x : al * h.x;
    h.y = acc.y * ri + bv.y; h.y = (h.y > 0.f) ? h.y : al * h.y;
    h.z = acc.z * ri + bv.z; h.z = (h.z > 0.f) ? h.z : al * h.z;
    h.w = acc.w * ri + bv.w; h.w = (h.w > 0.f) ? h.w : al * h.w;
    *(float4*)(out + (size_t)r * DFEAT + c4) = h;
}

// ---------------- column stats over h (block-local, 2 atomics/col/block) ----------------
__global__ void stats_kernel(const float* __restrict__ h,
                             float* __restrict__ sum, float* __restrict__ sq,
                             int n, int rowsPerBlock) {
    int d = threadIdx.x;                    // blockDim = 128
    int r0 = blockIdx.x * rowsPerBlock;
    if (r0 >= n) return;
    int rend = r0 + rowsPerBlock; if (rend > n) rend = n;
    float s = 0.0f, s2 = 0.0f;
    for (int r = r0; r < rend; ++r) {
        float v = h[(size_t)r * DFEAT + d];
        s += v;
        s2 += v * v;
    }
    atomicAdd(&sum[d], s);
    atomicAdd(&sq[d], s2);
}

// ---------------- fold BN stats into per-column scale/shift ----------------
__global__ void bnparams_kernel(const float* __restrict__ sum, const float* __restrict__ sq,
                                const float* __restrict__ gamma, const float* __restrict__ beta,
                                float* __restrict__ scale, float* __restrict__ shift, int n) {
    int d = threadIdx.x;
    float invN = 1.0f / (float)n;
    float mean = sum[d] * invN;
    float var = sq[d] * invN - mean * mean;
    float sc = gamma[d] * rsqrtf(var + 1e-5f);
    scale[d] = sc;
    shift[d] = beta[d] - mean * sc;
}

// ---------------- apply BN + outer PReLU(a2), in place on d_out ----------------
__global__ void bnapply_kernel(float* __restrict__ h, const float* __restrict__ scale,
                               const float* __restrict__ shift, const float* __restrict__ a2,
                               size_t total) {
    size_t i = (size_t)blockIdx.x * blockDim.x + threadIdx.x;
    if (i >= total) return;
    int d = (int)(i & (DFEAT - 1));
    float al = a2[0];
    float v = h[i] * scale[d] + shift[d];
    h[i] = (v > 0.0f) ? v : al * v;
}

extern "C" void kernel_launch(void* const* d_in, const int* in_sizes, int n_in,
                              void* d_out, int out_size, void* d_ws, size_t ws_size,
                              hipStream_t stream) {
    const float* heat  = (const float*)d_in[0];
    const float* W     = (const float*)d_in[1];
    const float* b     = (const float*)d_in[2];
    const float* a1    = (const float*)d_in[3];
    const float* gamma = (const float*)d_in[4];
    const float* beta  = (const float*)d_in[5];
    const float* a2    = (const float*)d_in[6];
    const int*   src   = (const int*)d_in[7];
    const int*   dst   = (const int*)d_in[8];

    const int n  = in_sizes[0] / DFEAT;   // 100000
    const int nE = in_sizes[7];           // 1600000
    const int nb = (n + 255) / 256;       // scan blocks (<=1024 supported)

    // workspace layout (4-byte words):
    // x[n*128] | degO[n] degI[n] cursor[n] sum[128] sq[128]  <- contiguous zero span
    // | rowPtr[n] | blockSum[1024] | scale[128] shift[128] | eSrc[nE] | Wp[16384]
    float*    x        = (float*)d_ws;
    unsigned* degO     = (unsigned*)(x + (size_t)n * DFEAT);
    unsigned* degI     = degO + n;
    unsigned* cursor   = degI + n;
    float*    sum      = (float*)(cursor + n);
    float*    sq       = sum + DFEAT;
    unsigned* rowPtr   = (unsigned*)(sq + DFEAT);
    unsigned* blockSum = rowPtr + n;
    float*    scale    = (float*)(blockSum + 1024);
    float*    shift    = scale + DFEAT;
    int*      eSrc     = (int*)(shift + DFEAT);
    float*    Wp       = (float*)(eSrc + nE);

    float* h = (float*)d_out;             // gather writes every row -> no d_out zeroing
    const size_t total = (size_t)n * DFEAT;

    // 1) zero degO|degI|cursor|sum|sq (one contiguous span)
    zero_kernel<<<512, 256, 0, stream>>>(degO, (size_t)3 * n + 2 * DFEAT);

    // 2) degrees
    degree_kernel<<<(nE + 255) / 256, 256, 0, stream>>>(src, dst, degO, degI, nE);

    // 3) exclusive scan degI -> rowPtr
    scanA_kernel<<<nb, 256, 0, stream>>>(degI, rowPtr, blockSum, n);
    scanB_kernel<<<1, 1024, 0, stream>>>(blockSum, nb);
    scanC_kernel<<<nb, 256, 0, stream>>>(rowPtr, blockSum, n);

    // 4) bucket fill edge lists
    fill_kernel<<<(nE + 255) / 256, 256, 0, stream>>>(src, dst, rowPtr, cursor, eSrc, nE);

    // 5) repack W for b64 B-tile loads
    repack_kernel<<<64, 256, 0, stream>>>(W, Wp);

    // 6) GEMM via fp32 WMMA: one wave per 16 rows
    {
        int waves = (n + 15) / 16;
        int blocks = (waves + 7) / 8;     // 256 threads = 8 waves
        gemm_wmma_kernel<<<blocks, 256, 0, stream>>>(heat, Wp, degO, x, n);
    }

    // 7) CSR gather + norm + bias + PReLU(a1): one wave per destination row
    {
        long long threads = (long long)n * 32;
        int blocks = (int)((threads + 255) / 256);
        gather_kernel<<<blocks, 256, 0, stream>>>(x, eSrc, rowPtr, degI, b, a1, h, n);
    }

    // 8) column stats
    {
        const int rowsPerBlock = 128;
        int blocks = (n + rowsPerBlock - 1) / rowsPerBlock;
        stats_kernel<<<blocks, DFEAT, 0, stream>>>(h, sum, sq, n, rowsPerBlock);
    }

    // 9) BN params
    bnparams_kernel<<<1, DFEAT, 0, stream>>>(sum, sq, gamma, beta, scale, shift, n);

    // 10) BN apply + outer PReLU(a2)
    {
        int blocks = (int)((total + 255) / 256);
        bnapply_kernel<<<blocks, 256, 0, stream>>>(h, scale, shift, a2, total);
    }
}